// Entangle_75136157876577
// MI455X (gfx1250) — compile-verified
//
#include <hip/hip_runtime.h>
#include <math.h>

typedef __attribute__((ext_vector_type(2))) float v2f;
typedef __attribute__((ext_vector_type(4))) float v4f;
typedef __attribute__((ext_vector_type(8))) float v8f;

#define NB 8
#define NS 16
#define NC 4
#define NN 256
#define NROW (NB*NS*NC)   /* 512 rows (b,s,c) */
#define SCN  (NS*NC)      /* 64 (s,c) pairs   */

/* workspace layout (float offsets), total ~3.07 MiB */
#define WS_SIGRE 0
#define WS_SIGIM (WS_SIGRE + NROW*NN)
#define WS_URE   (WS_SIGIM + NROW*NN)
#define WS_UIM   (WS_URE   + NROW*NN)
#define WS_VRE   (WS_UIM   + NROW*NN)
#define WS_VIM   (WS_VRE   + NROW*NN)
#define WS_MIXBAR (WS_VIM  + NROW*NN)

#define PI_OVER_128 0.02454369260617026f
#define LDSP 260   /* padded LDS row stride (floats): 16B aligned, bank spread 4/row */

/* ---------------- Kernel A: DFT via WMMA f32 -----------------------------
   sigRe[r,k] = sum_n x[r,n] cos(2pi nk/256); sigIm = -sum_n x[r,n] sin(...)
   One wave per 16(row)x16(k) tile; K-loop n in steps of 4.                 */
__global__ void dft_wmma(const float* __restrict__ x, float* __restrict__ ws) {
  const int lane = threadIdx.x;
  const int half = lane >> 4;
  const int lm   = lane & 15;
  const int kTile = blockIdx.x;     // 0..15
  const int rTile = blockIdx.y;     // 0..31
  const int rowA = rTile*16 + lm;   // A-frag: M = lane%16
  const int kout = kTile*16 + lm;   // B-frag: N = lane%16
  v8f accRe = {};
  v8f accIm = {};
  for (int n0 = 0; n0 < NN; n0 += 4) {
    const int nA = n0 + 2*half;     // A/B frag K indices: nA, nA+1
    v2f a;
    a.x = x[rowA*NN + nA];
    a.y = x[rowA*NN + nA + 1];
    v2f bc = {}, bs = {};
    {
      float sv, cv;
      __sincosf((float)((nA*kout) & 255) * PI_OVER_128, &sv, &cv);
      bc.x = cv; bs.x = -sv;
      __sincosf((float)(((nA+1)*kout) & 255) * PI_OVER_128, &sv, &cv);
      bc.y = cv; bs.y = -sv;
    }
    accRe = __builtin_amdgcn_wmma_f32_16x16x4_f32(false, a, false, bc, (short)0, accRe, false, false);
    accIm = __builtin_amdgcn_wmma_f32_16x16x4_f32(false, a, false, bs, (short)0, accIm, false, false);
  }
  float* sigRe = ws + WS_SIGRE;
  float* sigIm = ws + WS_SIGIM;
  for (int v = 0; v < 8; ++v) {
    const int r = rTile*16 + v + 8*half;   // C-layout: VGPR v -> M = v (+8 hi half)
    sigRe[r*NN + kout] = accRe[v];
    sigIm[r*NN + kout] = accIm[v];
  }
}

/* ---------------- Kernel B: corr / mix / mixbar (tiny) ------------------ */
__global__ void corr_mix(const float* __restrict__ sigRe, const float* __restrict__ gm,
                         const float* __restrict__ gs, float* __restrict__ mixbar) {
  __shared__ float shmix[NS][NS];
  const int tid = threadIdx.x;
  const int s1 = tid >> 4, s2 = tid & 15;
  float acc = 0.f;
  for (int b = 0; b < NB; ++b)
    for (int c = 0; c < NC; ++c) {
      const float t1 = sigRe[(b*SCN + s1*NC + c)*NN];  // DC term = sum(x)
      const float t2 = sigRe[(b*SCN + s2*NC + c)*NN];
      acc += t1*t2;
    }
  const float corr = acc * (1.0f/8192.0f);             // /(B*C*N)
  const float z = (corr - gm[s1]) / gs[s1];
  shmix[s1][s2] = expf(-0.5f*z*z);
  __syncthreads();
  if (tid < NS) {
    float m = 0.f;
    for (int j = 0; j < NS; ++j) m += shmix[tid][j];
    mixbar[tid] = m * (1.0f/16.0f);
  }
}

/* ---------------- Kernel C: U = M.sig, V = M^T.sig (complex) ------------
   One block per (s,c). Mask staged to LDS with double-buffered
   GLOBAL_LOAD_ASYNC_TO_LDS_B128 (ASYNCcnt-tracked), 8-row chunks.         */

#define ISSUE_CHUNK(chunk, buf)                                                \
  do {                                                                         \
    const int _i0 = (chunk)*8;                                                 \
    for (int _p = 0; _p < 2; ++_p) {                                           \
      const int _ii = _p*4 + (tid >> 6);                                       \
      const int _c4 = (tid & 63) * 4;                                          \
      const unsigned _g  = (unsigned)(((_i0 + _ii)*NN + _c4) * 4);             \
      const unsigned _lr = (unsigned)(size_t)&shMr[(buf)][_ii*LDSP + _c4];     \
      const unsigned _li = (unsigned)(size_t)&shMi[(buf)][_ii*LDSP + _c4];     \
      asm volatile("global_load_async_to_lds_b128 %0, %1, %2"                  \
                   :: "v"(_lr), "v"(_g), "s"(mR) : "memory");                  \
      asm volatile("global_load_async_to_lds_b128 %0, %1, %2"                  \
                   :: "v"(_li), "v"(_g), "s"(mI) : "memory");                  \
    }                                                                          \
  } while (0)

__global__ void matvec_uv(const float* __restrict__ maskR, const float* __restrict__ maskI,
                          float* __restrict__ ws) {
  __shared__ float shA[NB][LDSP];
  __shared__ float shB[NB][LDSP];
  __shared__ __align__(16) float shMr[2][8*LDSP];
  __shared__ __align__(16) float shMi[2][8*LDSP];
  __shared__ float shRed[2][64][4];
  const int tid = threadIdx.x;
  const int sc  = blockIdx.x;
  const float* sigRe = ws + WS_SIGRE;
  const float* sigIm = ws + WS_SIGIM;
  for (int b = 0; b < NB; ++b) {
    shA[b][tid] = sigRe[(b*SCN + sc)*NN + tid];
    shB[b][tid] = sigIm[(b*SCN + sc)*NN + tid];
  }
  const float* mR = maskR + (size_t)sc*NN*NN;
  const float* mI = maskI + (size_t)sc*NN*NN;

  float Vre[NB], Vim[NB];
  for (int b = 0; b < NB; ++b) { Vre[b] = 0.f; Vim[b] = 0.f; }

  ISSUE_CHUNK(0, 0);                               /* prologue */

  for (int chunk = 0; chunk < 32; ++chunk) {
    const int buf = chunk & 1;
    if (chunk < 31) {
      ISSUE_CHUNK(chunk + 1, buf ^ 1);
      asm volatile("s_wait_asynccnt 0x4" ::: "memory");  /* oldest 4 (this chunk) done */
    } else {
      asm volatile("s_wait_asynccnt 0x0" ::: "memory");
    }
    __syncthreads();
    const int i0 = chunk*8;
    /* V: thread owns column j = tid, accumulate over 8 rows of this chunk */
    for (int ii = 0; ii < 8; ++ii) {
      const float mr = shMr[buf][ii*LDSP + tid];
      const float mi = shMi[buf][ii*LDSP + tid];
      const int i = i0 + ii;
      for (int b = 0; b < NB; ++b) {
        const float Ai = shA[b][i], Bi = shB[b][i];
        Vre[b] += mr*Ai - mi*Bi;
        Vim[b] += mr*Bi + mi*Ai;
      }
    }
    /* U: (ii,b,quarter) partial sums over j, rotated start -> bank-free */
    {
      const int g = tid >> 2, quarter = tid & 3;
      const int uii = g >> 3, ub = g & 7;
      float ur = 0.f, ui = 0.f;
      for (int stp = 0; stp < 64; ++stp) {
        const int j = quarter*64 + ((stp + quarter) & 63);
        const float mr = shMr[buf][uii*LDSP + j];
        const float mi = shMi[buf][uii*LDSP + j];
        const float Aj = shA[ub][j], Bj = shB[ub][j];
        ur += mr*Aj - mi*Bj;
        ui += mr*Bj + mi*Aj;
      }
      shRed[0][g][quarter] = ur;
      shRed[1][g][quarter] = ui;
    }
    __syncthreads();
    if (tid < 64) {
      const float u0 = shRed[0][tid][0] + shRed[0][tid][1] + shRed[0][tid][2] + shRed[0][tid][3];
      const float u1 = shRed[1][tid][0] + shRed[1][tid][1] + shRed[1][tid][2] + shRed[1][tid][3];
      const int wii = tid >> 3, wb = tid & 7;
      ws[WS_URE + (wb*SCN + sc)*NN + i0 + wii] = u0;
      ws[WS_UIM + (wb*SCN + sc)*NN + i0 + wii] = u1;
    }
    __syncthreads();                               /* buf reusable next-next issue */
  }
  for (int b = 0; b < NB; ++b) {
    ws[WS_VRE + (b*SCN + sc)*NN + tid] = Vre[b];
    ws[WS_VIM + (b*SCN + sc)*NN + tid] = Vim[b];
  }
}

/* ---------------- Kernel D: s = Re(sig_i sig_j mask) — 128 MiB write ----
   b128 NT loads of mask (last use) and b128 NT stores of the streamed
   output so the L2 keeps useful data. Mask tile reused for all 8 batches. */
__global__ void sup_out(const float* __restrict__ maskR, const float* __restrict__ maskI,
                        const float* __restrict__ ws, float* __restrict__ outS) {
  __shared__ float shA[NB][LDSP];
  __shared__ float shB[NB][LDSP];
  const int tid = threadIdx.x;
  const int sc  = blockIdx.x;
  const float* sigRe = ws + WS_SIGRE;
  const float* sigIm = ws + WS_SIGIM;
  for (int b = 0; b < NB; ++b) {
    shA[b][tid] = sigRe[(b*SCN + sc)*NN + tid];
    shB[b][tid] = sigIm[(b*SCN + sc)*NN + tid];
  }
  __syncthreads();
  const int j0 = (tid & 63) * 4;      /* thread owns 4 consecutive j */
  const int io = tid >> 6;            /* 4 rows processed per step   */
  float Ajv[NB][4], Bjv[NB][4];
  for (int b = 0; b < NB; ++b)
    for (int q = 0; q < 4; ++q) { Ajv[b][q] = shA[b][j0+q]; Bjv[b][q] = shB[b][j0+q]; }
  const float* mR = maskR + (size_t)sc*NN*NN;
  const float* mI = maskI + (size_t)sc*NN*NN;
  for (int ib = 0; ib < NN; ib += 4) {
    const int i = ib + io;
    const v4f mr = __builtin_nontemporal_load((const v4f*)(mR + i*NN + j0));
    const v4f mi = __builtin_nontemporal_load((const v4f*)(mI + i*NN + j0));
    for (int b = 0; b < NB; ++b) {
      const float Ai = shA[b][i], Bi = shB[b][i];    /* LDS broadcast */
      v4f o;
      for (int q = 0; q < 4; ++q) {
        const float tr = Ai*Ajv[b][q] - Bi*Bjv[b][q];  /* Re(sig_i sig_j) */
        const float ti = Ai*Bjv[b][q] + Bi*Ajv[b][q];  /* Im(sig_i sig_j) */
        o[q] = tr*mr[q] - ti*mi[q];
      }
      __builtin_nontemporal_store(o, (v4f*)(outS + ((size_t)(b*SCN + sc)*NN + i)*NN + j0));
    }
  }
}

/* ---------------- Kernel E: weighted iDFT (irfft) via WMMA + y blend ----
   rowsum = sig.*U, colsum = sig.*V built on the fly as A-fragments.
   Basis carries irfft weights: w0=wNyq=1/N, else 2/N; Ci = -w sin.        */
__global__ void irfft_y(const float* __restrict__ x, const float* __restrict__ pol,
                        const float* __restrict__ ws, float* __restrict__ outY) {
  const int lane = threadIdx.x;
  const int half = lane >> 4;
  const int lm   = lane & 15;
  const int tTile = blockIdx.x;     // 0..15
  const int rTile = blockIdx.y;     // 0..31
  const int rowA = rTile*16 + lm;
  const int tout = tTile*16 + lm;
  const float* sigRe = ws + WS_SIGRE;
  const float* sigIm = ws + WS_SIGIM;
  const float* Ure = ws + WS_URE;
  const float* Uim = ws + WS_UIM;
  const float* Vre = ws + WS_VRE;
  const float* Vim = ws + WS_VIM;
  const float* mixbar = ws + WS_MIXBAR;
  v8f accR = {};
  v8f accC = {};
  for (int k0 = 0; k0 <= 128; k0 += 4) {        // 33 steps; w=0 past Nyquist
    const int kA = k0 + 2*half;
    v2f Rr = {}, Ri = {}, Sr = {}, Si = {}, Cr = {}, Ci = {};
    for (int t = 0; t < 2; ++t) {
      const int k = kA + t;
      const int idx = rowA*NN + k;
      const float A = sigRe[idx], B = sigIm[idx];
      const float ur = Ure[idx], ui = Uim[idx];
      const float vr = Vre[idx], vi = Vim[idx];
      const float rr = A*ur - B*ui;             // Re(sig*U)
      const float ri = A*ui + B*ur;             // Im(sig*U)
      const float sr = A*vr - B*vi;             // Re(sig*V)
      const float si = A*vi + B*vr;             // Im(sig*V)
      const float w = (k == 0 || k == 128) ? (1.0f/256.0f)
                     : (k < 128 ? (2.0f/256.0f) : 0.0f);
      float sv, cv;
      __sincosf((float)((k*tout) & 255) * PI_OVER_128, &sv, &cv);
      if (t == 0) { Rr.x = rr; Ri.x = ri; Sr.x = sr; Si.x = si; Cr.x = w*cv; Ci.x = -w*sv; }
      else        { Rr.y = rr; Ri.y = ri; Sr.y = sr; Si.y = si; Cr.y = w*cv; Ci.y = -w*sv; }
    }
    accR = __builtin_amdgcn_wmma_f32_16x16x4_f32(false, Rr, false, Cr, (short)0, accR, false, false);
    accR = __builtin_amdgcn_wmma_f32_16x16x4_f32(false, Ri, false, Ci, (short)0, accR, false, false);
    accC = __builtin_amdgcn_wmma_f32_16x16x4_f32(false, Sr, false, Cr, (short)0, accC, false, false);
    accC = __builtin_amdgcn_wmma_f32_16x16x4_f32(false, Si, false, Ci, (short)0, accC, false, false);
  }
  for (int v = 0; v < 8; ++v) {
    const int r = rTile*16 + v + 8*half;
    const int s = (r >> 2) & 15;                // r = b*64 + s*4 + c
    const float mb = mixbar[s];
    float sp, cp;
    __sincosf(pol[s], &sp, &cp);
    const float smear = cp*accR[v] + sp*accC[v];
    outY[r*NN + tout] = mb*smear + (1.0f - mb)*x[r*NN + tout];
  }
}

extern "C" void kernel_launch(void* const* d_in, const int* in_sizes, int n_in,
                              void* d_out, int out_size, void* d_ws, size_t ws_size,
                              hipStream_t stream) {
  (void)in_sizes; (void)n_in; (void)out_size; (void)ws_size;
  const float* x   = (const float*)d_in[0];
  const float* mr  = (const float*)d_in[1];
  const float* mi  = (const float*)d_in[2];
  const float* pol = (const float*)d_in[3];
  const float* gm  = (const float*)d_in[4];
  const float* gs  = (const float*)d_in[5];
  float* out  = (float*)d_out;
  float* ws   = (float*)d_ws;            /* needs ~3.1 MiB */
  float* outY = out;                     /* 131072 floats   */
  float* outS = out + NROW*NN;           /* 33554432 floats */

  dft_wmma <<<dim3(16,32), dim3(32), 0, stream>>>(x, ws);
  corr_mix <<<1, 256, 0, stream>>>(ws + WS_SIGRE, gm, gs, ws + WS_MIXBAR);
  matvec_uv<<<SCN, 256, 0, stream>>>(mr, mi, ws);   /* warms L2 with mask  */
  sup_out  <<<SCN, 256, 0, stream>>>(mr, mi, ws, outS);
  irfft_y  <<<dim3(16,32), dim3(32), 0, stream>>>(x, pol, ws, outY);
}